// Recommender_47064251629718
// MI455X (gfx1250) — compile-verified
//
#include <hip/hip_runtime.h>
#include <hip/hip_bf16.h>
#include <math.h>

#define NB    4096
#define LH    200
#define DE    64
#define XDIM  320   // 4*DE + DE
#define H1DIM 128
#define H2DIM 64

typedef __attribute__((ext_vector_type(2))) float v2f;
typedef __attribute__((ext_vector_type(8))) float v8f;

// ---------- order-preserving float <-> uint mapping for radix select ----------
__device__ __forceinline__ unsigned f2k(float f) {
    unsigned u = __float_as_uint(f);
    return (u & 0x80000000u) ? ~u : (u | 0x80000000u);
}
__device__ __forceinline__ float k2f(unsigned k) {
    unsigned u = (k & 0x80000000u) ? (k ^ 0x80000000u) : ~k;
    return __uint_as_float(u);
}

// =====================================================================
// Kernel 1: per-user weighted-embedding stats (min/max/mean/median),
// L2-normalize, concat with film embedding -> X[b][0..319] in workspace.
// One block per user; 64 threads (= D lanes); 2 waves.
// =====================================================================
__global__ __launch_bounds__(64)
void user_feat_kernel(const int* __restrict__ film_hist,
                      const float* __restrict__ ratings,
                      const int* __restrict__ lengths,
                      const int* __restrict__ film_indices,
                      const float* __restrict__ emb_table,
                      float* __restrict__ X)
{
    __shared__ unsigned keys[LH * DE];   // 51.2 KB; bank(l*64+d) = d -> conflict-free
    __shared__ float    wts[LH];
    __shared__ float    red[64];
    __shared__ float    ue[4 * DE];

    const int b   = blockIdx.x;
    const int tid = threadIdx.x;         // == d lane
    const int n   = lengths[b];

    // ---- mean rating over valid entries ----
    float s = 0.0f;
    for (int l = tid; l < n; l += 64) s += ratings[b * LH + l];
    red[tid] = s;
    __syncthreads();
    for (int off = 32; off > 0; off >>= 1) {
        if (tid < off) red[tid] += red[tid + off];
        __syncthreads();
    }
    const float inv_n  = 1.0f / (float)n;
    const float mean_r = red[0] * inv_n;
    __syncthreads();

    // ---- weights ----
    for (int l = tid; l < n; l += 64) wts[l] = ratings[b * LH + l] - mean_r + 0.1f;
    __syncthreads();

    // ---- gather embedding rows, form w = e*weight, track min/max/sum, stash keys ----
    float vmin = INFINITY, vmax = -INFINITY, vsum = 0.0f;
    for (int l = 0; l < n; ++l) {
        const int   f = film_hist[b * LH + l];          // uniform -> scalar load
        const float e = emb_table[(size_t)f * DE + tid];// 256B coalesced row
        const float w = e * wts[l];
        vmin = fminf(vmin, w);
        vmax = fmaxf(vmax, w);
        vsum += w;
        keys[l * DE + tid] = f2k(w);
    }
    __syncthreads();

    // ---- median via 4-bit radix select over own LDS column ----
    unsigned rank   = (unsigned)((n - 1) >> 1);
    unsigned prefix = 0;
    for (int shift = 28; shift >= 0; shift -= 4) {
        const int hs = shift + 4;
        unsigned long long c0 = 0ull, c1 = 0ull;     // 16 byte-wide bucket counters
        for (int l = 0; l < n; ++l) {
            const unsigned k = keys[l * DE + tid];
            const bool match = (hs >= 32) ? true : ((k >> hs) == (prefix >> hs));
            if (match) {
                const unsigned bk = (k >> shift) & 15u;
                if (bk < 8u) c0 += 1ull << (bk * 8u);
                else         c1 += 1ull << ((bk - 8u) * 8u);
            }
        }
        unsigned bk = 0;
        for (; bk < 16u; ++bk) {
            const unsigned cnt = (bk < 8u)
                ? (unsigned)((c0 >> (bk * 8u)) & 0xFFull)
                : (unsigned)((c1 >> ((bk - 8u) * 8u)) & 0xFFull);
            if (rank < cnt) break;
            rank -= cnt;
        }
        prefix |= (bk << shift);
    }
    const float vmed = k2f(prefix);

    ue[tid]           = vmin;
    ue[DE + tid]      = vmax;
    ue[2 * DE + tid]  = vsum * inv_n;
    ue[3 * DE + tid]  = vmed;
    __syncthreads();

    // ---- L2 norm over 256 values ----
    float ss = 0.0f;
    for (int i = tid; i < 4 * DE; i += 64) ss += ue[i] * ue[i];
    red[tid] = ss;
    __syncthreads();
    for (int off = 32; off > 0; off >>= 1) {
        if (tid < off) red[tid] += red[tid + off];
        __syncthreads();
    }
    const float rinv = rsqrtf(red[0]);

    // ---- emit feature row ----
    float* xrow = X + (size_t)b * XDIM;
    for (int i = tid; i < 4 * DE; i += 64) xrow[i] = ue[i] * rinv;
    xrow[4 * DE + tid] = emb_table[(size_t)film_indices[b] * DE + tid];
}

// =====================================================================
// Kernel 2: MLP via V_WMMA_F32_16X16X4_F32. One wave per 16-user tile.
// =====================================================================
__device__ __forceinline__ v8f wmma_f32_4(v2f a, v2f b, v8f c) {
#if defined(__gfx1250__)
    return __builtin_amdgcn_wmma_f32_16x16x4_f32(false, a, false, b, (short)0, c,
                                                 false, false);
#else
    return c; // host-pass stub; never executed
#endif
}

#define XS_STR  324   // 320 padded; stride%64==4 -> bank-conflict-free A reads
#define H1_STR  132
#define H2_STR  68

__global__ __launch_bounds__(32)
void mlp_wmma_kernel(const float* __restrict__ X,
                     const float* __restrict__ W1, const float* __restrict__ b1,
                     const float* __restrict__ W2, const float* __restrict__ b2,
                     const float* __restrict__ W3, const float* __restrict__ b3,
                     float* __restrict__ out)
{
    __shared__ float xs[16 * XS_STR];
    __shared__ float h1s[16 * H1_STR];
    __shared__ float h2s[16 * H2_STR];

    const int lane = threadIdx.x;        // 0..31 (wave32)
    const int b0   = blockIdx.x * 16;

    // X tile is one contiguous 16*320 chunk -> coalesced load into padded LDS
    for (int i = lane; i < 16 * XDIM; i += 32) {
        xs[(i / XDIM) * XS_STR + (i % XDIM)] = X[(size_t)b0 * XDIM + i];
    }
    __syncthreads();

    const int m    = lane & 15;                 // A-frag row / C-frag column
    const int klo  = (lane < 16) ? 0 : 2;       // K slot pair for this half-wave
    const int hrow = (lane < 16) ? 0 : 8;       // C-frag M offset

    // ---------- Layer 1: (16x320) @ (320x128) + b1, ReLU ----------
    for (int nt = 0; nt < 8; ++nt) {
        const int   nc   = nt * 16 + m;
        const float bias = b1[nc];
        v8f acc = {bias, bias, bias, bias, bias, bias, bias, bias};
        for (int k0 = 0; k0 < XDIM; k0 += 4) {
            v2f a, bb;
            a.x  = xs[m * XS_STR + k0 + klo];
            a.y  = xs[m * XS_STR + k0 + klo + 1];
            bb.x = W1[(k0 + klo) * H1DIM + nc];
            bb.y = W1[(k0 + klo + 1) * H1DIM + nc];
            acc  = wmma_f32_4(a, bb, acc);
        }
#pragma unroll
        for (int r = 0; r < 8; ++r) {
            const float v = acc[r];
            h1s[(r + hrow) * H1_STR + nc] = v > 0.0f ? v : 0.0f;
        }
    }
    __syncthreads();

    // ---------- Layer 2: (16x128) @ (128x64) + b2, ReLU ----------
    for (int nt = 0; nt < 4; ++nt) {
        const int   nc   = nt * 16 + m;
        const float bias = b2[nc];
        v8f acc = {bias, bias, bias, bias, bias, bias, bias, bias};
        for (int k0 = 0; k0 < H1DIM; k0 += 4) {
            v2f a, bb;
            a.x  = h1s[m * H1_STR + k0 + klo];
            a.y  = h1s[m * H1_STR + k0 + klo + 1];
            bb.x = W2[(k0 + klo) * H2DIM + nc];
            bb.y = W2[(k0 + klo + 1) * H2DIM + nc];
            acc  = wmma_f32_4(a, bb, acc);
        }
#pragma unroll
        for (int r = 0; r < 8; ++r) {
            const float v = acc[r];
            h2s[(r + hrow) * H2_STR + nc] = v > 0.0f ? v : 0.0f;
        }
    }
    __syncthreads();

    // ---------- Layer 3: (16x64) @ (64x1) + b3, sigmoid ----------
    if (lane < 16) {
        float sacc = b3[0];
#pragma unroll 8
        for (int k = 0; k < H2DIM; ++k) sacc += h2s[lane * H2_STR + k] * W3[k];
        out[b0 + lane] = 1.0f / (1.0f + expf(-sacc));
    }
}

// =====================================================================
extern "C" void kernel_launch(void* const* d_in, const int* in_sizes, int n_in,
                              void* d_out, int out_size, void* d_ws, size_t ws_size,
                              hipStream_t stream)
{
    const int*   film_hist    = (const int*)  d_in[0];
    const float* ratings      = (const float*)d_in[1];
    const int*   lengths      = (const int*)  d_in[2];
    const int*   film_indices = (const int*)  d_in[3];
    const float* emb_table    = (const float*)d_in[4];
    const float* W1           = (const float*)d_in[5];
    const float* b1           = (const float*)d_in[6];
    const float* W2           = (const float*)d_in[7];
    const float* b2           = (const float*)d_in[8];
    const float* W3           = (const float*)d_in[9];
    const float* b3           = (const float*)d_in[10];
    float*       out          = (float*)d_out;

    float* X = (float*)d_ws;   // (NB, 320) f32 feature rows: 5.24 MB

    user_feat_kernel<<<NB, 64, 0, stream>>>(film_hist, ratings, lengths,
                                            film_indices, emb_table, X);
    mlp_wmma_kernel<<<NB / 16, 32, 0, stream>>>(X, W1, b1, W2, b2, W3, b3, out);
}